// EngramCodebook_40192303956596
// MI455X (gfx1250) — compile-verified
//
#include <hip/hip_runtime.h>
#include <hip/hip_bf16.h>
#include <stdint.h>

#define N_TOK 16384
#define D 2048
#define NSEED 4096

typedef __attribute__((ext_vector_type(2))) float v2f;
typedef __attribute__((ext_vector_type(8))) float v8f;

struct F2 { float x, y; };
struct __align__(16) F4 { float x, y, z, w; };

// ---------------------------------------------------------------- init key
__global__ void k_init(unsigned long long* key) {
    if (threadIdx.x == 0 && blockIdx.x == 0) *key = ~0ull;
}

// ------------------------------------------- column partial sums (coalesced)
// grid (D/256, 16), block 256: partial[y][col] = sum over 1024 rows
__global__ void k_pool_partial(const float* __restrict__ hs,
                               float* __restrict__ partial) {
    int col = blockIdx.x * 256 + threadIdx.x;
    int r0  = blockIdx.y * (N_TOK / 16);
    const float* p = hs + (size_t)r0 * D + col;
    float s = 0.f;
    for (int r = 0; r < N_TOK / 16; ++r) {
        if ((r & 31) == 0) __builtin_prefetch(p + (size_t)32 * D, 0, 0);
        s += *p;
        p += D;
    }
    partial[(size_t)blockIdx.y * D + col] = s;
}

// ------------------------------- fixed-order reduce -> deterministic pooled
__global__ void k_pool_reduce(const float* __restrict__ partial,
                              float* __restrict__ pooled) {
    int col = blockIdx.x * 256 + threadIdx.x;
    float s = 0.f;
    for (int y = 0; y < 16; ++y) s += partial[(size_t)y * D + col];
    pooled[col] = s * (1.0f / (float)N_TOK);
}

// --------------------------- seed distances via V_WMMA_F32_16X16X4_F32 (fp32)
// one wave32 per block, 16 seeds per wave; score = ||s||^2 - 2*(s.p)
__global__ void __launch_bounds__(32)
k_dist(const float* __restrict__ seed, const float* __restrict__ pooled,
       unsigned long long* __restrict__ key) {
    __shared__ float lds_nrm[32];
    __shared__ float lds_dot[16];

    const int lane = threadIdx.x;
    const int m    = lane & 15;
    const int off  = (lane < 16) ? 0 : 2;   // K offset per ISA A/B f32 layout
    const int seedBase = blockIdx.x * 16;
    const float* srow = seed + (size_t)(seedBase + m) * D;

    v8f c = {};
    float nrm = 0.f;
    for (int k = 0; k < D; k += 4) {
        F2 av = *(const F2*)(srow   + k + off);   // A: 16x4 f32 tile
        F2 bv = *(const F2*)(pooled + k + off);   // B: 4x16, all columns = pooled
        v2f a; a[0] = av.x; a[1] = av.y;
        v2f b; b[0] = bv.x; b[1] = bv.y;
        c = __builtin_amdgcn_wmma_f32_16x16x4_f32(
                /*neg_a=*/false, a, /*neg_b=*/false, b,
                /*c_mod=*/(short)0, c, /*reuse_a=*/false, /*reuse_b=*/false);
        nrm = __builtin_fmaf(av.x, av.x, nrm);    // co-executes with XDL wmma
        nrm = __builtin_fmaf(av.y, av.y, nrm);
    }

    lds_nrm[lane] = nrm;
    if (lane == 0) {
        #pragma unroll
        for (int r = 0; r < 8; ++r) lds_dot[r] = c[r];       // seeds 0..7
    }
    if (lane == 16) {
        #pragma unroll
        for (int r = 0; r < 8; ++r) lds_dot[8 + r] = c[r];   // seeds 8..15
    }
    __syncthreads();

    if (lane == 0) {
        float best = 3.4e38f; int bestIdx = seedBase;
        for (int i = 0; i < 16; ++i) {
            float score = (lds_nrm[i] + lds_nrm[i + 16]) - 2.0f * lds_dot[i];
            if (score < best) { best = score; bestIdx = seedBase + i; }
        }
        unsigned u = __float_as_uint(best);
        u = (u & 0x80000000u) ? ~u : (u | 0x80000000u);      // monotone map
        unsigned long long k64 =
            ((unsigned long long)u << 32) | (unsigned)bestIdx;
        atomicMin(key, k64);                                  // ties -> lowest idx
    }
}

// ------------------------------------------ usage(+1 at idx) and idx scalar
__global__ void k_usage(const float* __restrict__ usage_in,
                        const unsigned long long* __restrict__ key,
                        float* __restrict__ out) {
    unsigned idx = (unsigned)(*key & 0xFFFFFFFFull);
    int j = blockIdx.x * 256 + threadIdx.x;
    float v = usage_in[j] + ((j == (int)idx) ? 1.0f : 0.0f);
    out[(size_t)N_TOK * D + 1 + j] = v;
    if (j == 0) out[(size_t)N_TOK * D] = (float)idx;
}

// ----------------------------------- broadcast recon: 128 MiB of B128 stores
__global__ void k_recon(const float* __restrict__ seed,
                        const unsigned long long* __restrict__ key,
                        float* __restrict__ out) {
    unsigned idx = (unsigned)(*key & 0xFFFFFFFFull);
    size_t flat = (size_t)blockIdx.x * 256 + threadIdx.x;   // float4 index
    unsigned colv = (unsigned)(flat & (D / 4 - 1));
    F4 v = ((const F4*)(seed + (size_t)idx * D))[colv];     // seed row (L2-hot)
    ((F4*)out)[flat] = v;
}

extern "C" void kernel_launch(void* const* d_in, const int* in_sizes, int n_in,
                              void* d_out, int out_size, void* d_ws, size_t ws_size,
                              hipStream_t stream) {
    const float* hs    = (const float*)d_in[0];
    const float* seed  = (const float*)d_in[1];
    const float* usage = (const float*)d_in[2];
    float* out = (float*)d_out;

    // d_ws layout: pooled[2048] floats, then 8-byte argmin key  (~8.2 KB)
    float* pooled = (float*)d_ws;
    unsigned long long* key =
        (unsigned long long*)((char*)d_ws + D * sizeof(float));
    // Big partial-sum scratch lives in the recon region of d_out (16*2048
    // floats); it is consumed by k_pool_reduce and fully overwritten by
    // k_recon at the end of this same launch.
    float* partial = out;

    k_init<<<1, 1, 0, stream>>>(key);
    k_pool_partial<<<dim3(D / 256, 16), 256, 0, stream>>>(hs, partial);
    k_pool_reduce<<<D / 256, 256, 0, stream>>>(partial, pooled);
    k_dist<<<NSEED / 16, 32, 0, stream>>>(seed, pooled, key);
    k_usage<<<NSEED / 256, 256, 0, stream>>>(usage, key, out);
    k_recon<<<((size_t)N_TOK * D) / (4 * 256), 256, 0, stream>>>(seed, key, out);
}